// TravelTime_16011638080279
// MI455X (gfx1250) — compile-verified
//
#include <hip/hip_runtime.h>
#include <math.h>

// ---------------- problem constants (match reference) ----------------
#define TPB   256          // 8 wave32 per block on CDNA5
#define PPT   16           // picks per thread (4 groups of 4 consecutive)
#define PPB   (TPB * PPT)

#define NUM_EVENT_C   10000
#define NUM_STATION_C 1000
#define NR_C          1500
#define NZ_C          400

constexpr float H_C    = 0.1f;
constexpr float RG0_C  = 0.0f;
constexpr float ZG0_C  = -5.0f;
constexpr float REG_DT = 0.1f;

typedef int   __attribute__((ext_vector_type(4))) i32x4;
typedef float __attribute__((ext_vector_type(4))) f32x4;

// ---------------- gfx1250 async global->LDS staging ----------------
#if defined(__has_builtin)
#  if __has_builtin(__builtin_amdgcn_global_load_async_to_lds_b32) && \
      __has_builtin(__builtin_amdgcn_s_wait_asynccnt)
#    define HAVE_ASYNC_LDS 1
#  endif
#  if __has_builtin(__builtin_amdgcn_global_load_async_to_lds_b128)
#    define HAVE_ASYNC_B128 1
#  endif
#endif

#if defined(HAVE_ASYNC_LDS)
typedef __attribute__((address_space(1))) int   gi32_t;
typedef __attribute__((address_space(3))) int   li32_t;
#endif
#if defined(HAVE_ASYNC_B128)
typedef __attribute__((address_space(1))) i32x4 gi32x4_t;
typedef __attribute__((address_space(3))) i32x4 li32x4_t;
#endif

__device__ __forceinline__ void stage_to_lds(const float* __restrict__ g,
                                             float* s, int n, int tid) {
#if defined(HAVE_ASYNC_B128) && defined(HAVE_ASYNC_LDS)
  const int n4 = n >> 2;  // tables are all multiples of 4 floats
  for (int i = tid; i < n4; i += TPB) {
    __builtin_amdgcn_global_load_async_to_lds_b128(
        (gi32x4_t*)((const i32x4*)g + i), (li32x4_t*)((i32x4*)s + i),
        /*offset=*/0, /*cpol=*/0);
  }
  for (int i = (n4 << 2) + tid; i < n; i += TPB) {  // tail (normally empty)
    __builtin_amdgcn_global_load_async_to_lds_b32(
        (gi32_t*)(g + i), (li32_t*)(s + i), 0, 0);
  }
#elif defined(HAVE_ASYNC_LDS)
  for (int i = tid; i < n; i += TPB) {
    __builtin_amdgcn_global_load_async_to_lds_b32(
        (gi32_t*)(g + i), (li32_t*)(s + i), 0, 0);
  }
#else
  for (int i = tid; i < n; i += TPB) s[i] = g[i];
#endif
}

__device__ __forceinline__ int clampi(int v, int lo, int hi) {
  return v < lo ? lo : (v > hi ? hi : v);
}

// one pick: gathers from LDS tables + L2 tables, bilinear interp, huber accum
__device__ __forceinline__ float compute_pick(
    int si, int ei, int pt, float ptm, float pw,
    const float* __restrict__ event_loc, const float* __restrict__ timetable,
    const float* s_stloc, const float* s_stdt, const float* s_evt,
    float& lloss) {
  const float sx  = s_stloc[si * 3 + 0];
  const float sy  = s_stloc[si * 3 + 1];
  const float sz  = s_stloc[si * 3 + 2];
  const float sdt = s_stdt [si * 2 + pt];
  const float et  = s_evt  [ei];

  const float ex = event_loc[ei * 3 + 0];
  const float ey = event_loc[ei * 3 + 1];
  const float ez = event_loc[ei * 3 + 2];

  const float dx = ex - sx, dy = ey - sy;
  const float r = sqrtf(dx * dx + dy * dy);
  const float z = ez - sz;

  // index math exactly as the reference (divide by H, clamp, cast)
  const int ir0 = clampi((int)floorf((r - RG0_C) / H_C), 0, NR_C - 2);
  const int iz0 = clampi((int)floorf((z - ZG0_C) / H_C), 0, NZ_C - 2);
  const float x1 = (float)ir0 * H_C + RG0_C;
  const float y1 = (float)iz0 * H_C + ZG0_C;
  const float x2 = x1 + H_C;
  const float y2 = y1 + H_C;

  const int b = pt * (NR_C * NZ_C) + ir0 * NZ_C + iz0;
  const float Q11 = timetable[b];
  const float Q12 = timetable[b + 1];
  const float Q21 = timetable[b + NZ_C];
  const float Q22 = timetable[b + NZ_C + 1];

  const float ttv = (Q11 * (x2 - r) * (y2 - z) + Q21 * (r - x1) * (y2 - z) +
                     Q12 * (x2 - r) * (z - y1) + Q22 * (r - x1) * (z - y1)) /
                    (H_C * H_C);

  const float t = et + ttv + sdt;
  const float e = t - ptm;
  const float a = fabsf(e);
  const float hub = (a < 1.0f) ? 0.5f * e * e : (a - 0.5f);
  lloss += hub * pw + REG_DT * fabsf(sdt);
  return t;
}

// ---------------- main kernel: t + per-block loss partials ----------------
__global__ __launch_bounds__(TPB) void tt_loss_kernel(
    const int*   __restrict__ station_index,
    const int*   __restrict__ event_index,
    const int*   __restrict__ phase_type,
    const float* __restrict__ phase_time,
    const float* __restrict__ phase_weight,
    const float* __restrict__ event_loc,     // [NE,3]
    const float* __restrict__ event_time,    // [NE]
    const float* __restrict__ station_loc,   // [NS,3]
    const float* __restrict__ station_dt,    // [NS,2]
    const float* __restrict__ timetable,     // [2, NR*NZ] (ir*NZ+iz)
    float* __restrict__ t_out,
    float* __restrict__ block_partials,
    int n) {
  // 60KB LDS: 5 WGs/WGP (320KB), 40 waves/WGP for gather-latency hiding.
  __shared__ __attribute__((aligned(16))) float s_stloc[NUM_STATION_C * 3];
  __shared__ __attribute__((aligned(16))) float s_stdt [NUM_STATION_C * 2];
  __shared__ __attribute__((aligned(16))) float s_evt  [NUM_EVENT_C];
  __shared__ float s_wred[TPB / 32];

  const int tid = threadIdx.x;
  stage_to_lds(station_loc, s_stloc, NUM_STATION_C * 3, tid);
  stage_to_lds(station_dt,  s_stdt,  NUM_STATION_C * 2, tid);
  stage_to_lds(event_time,  s_evt,   NUM_EVENT_C,       tid);
#if defined(HAVE_ASYNC_LDS)
  __builtin_amdgcn_s_wait_asynccnt(0);
#endif
  __syncthreads();

  // each thread handles PPT/4 groups of 4 consecutive picks -> b128 NT streams
  const int vbase = blockIdx.x * (PPB / 4) + tid;
  float lloss = 0.0f;

#pragma unroll
  for (int g = 0; g < PPT / 4; ++g) {
    const int v    = vbase + g * TPB;  // vec4 index
    const int idx0 = v * 4;
    if (idx0 + 3 < n) {
      const i32x4 si4  = __builtin_nontemporal_load((const i32x4*)station_index + v);
      const i32x4 ei4  = __builtin_nontemporal_load((const i32x4*)event_index   + v);
      const i32x4 pt4  = __builtin_nontemporal_load((const i32x4*)phase_type    + v);
      const f32x4 ptm4 = __builtin_nontemporal_load((const f32x4*)phase_time    + v);
      const f32x4 pw4  = __builtin_nontemporal_load((const f32x4*)phase_weight  + v);
      f32x4 t4;
#pragma unroll
      for (int k = 0; k < 4; ++k) {
        t4[k] = compute_pick(si4[k], ei4[k], pt4[k], ptm4[k], pw4[k],
                             event_loc, timetable, s_stloc, s_stdt, s_evt, lloss);
      }
      __builtin_nontemporal_store(t4, (f32x4*)t_out + v);
    } else {
#pragma unroll
      for (int k = 0; k < 4; ++k) {
        const int idx = idx0 + k;
        if (idx < n) {
          const int   si  = station_index[idx];
          const int   ei  = event_index[idx];
          const int   pt  = phase_type[idx];
          const float ptm = phase_time[idx];
          const float pw  = phase_weight[idx];
          const float t = compute_pick(si, ei, pt, ptm, pw, event_loc, timetable,
                                       s_stloc, s_stdt, s_evt, lloss);
          t_out[idx] = t;
        }
      }
    }
  }

  // wave32 reduction (CDNA5: 8 waves per 256-thread block)
  for (int off = 16; off > 0; off >>= 1)
    lloss += __shfl_down(lloss, off, 32);
  const int lane = tid & 31;
  const int wv   = tid >> 5;
  if (lane == 0) s_wred[wv] = lloss;
  __syncthreads();
  if (tid == 0) {
    float s = 0.0f;
#pragma unroll
    for (int w = 0; w < TPB / 32; ++w) s += s_wred[w];
    block_partials[blockIdx.x] = s;  // deterministic: fixed order, no atomics
  }
}

// ---------------- deterministic final reduction ----------------
__global__ __launch_bounds__(TPB) void tt_reduce_kernel(
    const float* __restrict__ partials, int nblocks, float* __restrict__ loss_out) {
  __shared__ float s[TPB];
  float acc = 0.0f;
  for (int i = threadIdx.x; i < nblocks; i += TPB) acc += partials[i];
  s[threadIdx.x] = acc;
  __syncthreads();
  for (int st = TPB / 2; st > 0; st >>= 1) {
    if (threadIdx.x < st) s[threadIdx.x] += s[threadIdx.x + st];
    __syncthreads();
  }
  if (threadIdx.x == 0) *loss_out = s[0];
}

extern "C" void kernel_launch(void* const* d_in, const int* in_sizes, int n_in,
                              void* d_out, int out_size, void* d_ws, size_t ws_size,
                              hipStream_t stream) {
  const int*   station_index = (const int*)  d_in[0];
  const int*   event_index   = (const int*)  d_in[1];
  const int*   phase_type    = (const int*)  d_in[2];
  const float* phase_time    = (const float*)d_in[3];
  const float* phase_weight  = (const float*)d_in[4];
  const float* event_loc     = (const float*)d_in[5];
  const float* event_time    = (const float*)d_in[6];
  const float* station_loc   = (const float*)d_in[7];
  const float* station_dt    = (const float*)d_in[8];
  const float* timetable     = (const float*)d_in[9];
  // d_in[10], d_in[11] = timetable gradients: unused by the forward reference.
  (void)n_in; (void)ws_size;

  const int n = in_sizes[0];
  float* t_out    = (float*)d_out;
  float* loss_out = ((float*)d_out) + (out_size - 1);
  float* partials = (float*)d_ws;

  const int blocks = (n + PPB - 1) / PPB;
  tt_loss_kernel<<<blocks, TPB, 0, stream>>>(
      station_index, event_index, phase_type, phase_time, phase_weight,
      event_loc, event_time, station_loc, station_dt, timetable,
      t_out, partials, n);
  tt_reduce_kernel<<<1, TPB, 0, stream>>>(partials, blocks, loss_out);
}